// Attention_40338332844097
// MI455X (gfx1250) — compile-verified
//
#include <hip/hip_runtime.h>
#include <hip/hip_bf16.h>

// ---------------------------------------------------------------------------
// Fused QKV projection + multi-head attention for MI455X (gfx1250, wave32).
// All matrix math uses V_WMMA_F32_16X16X4_F32 (full fp32 precision; the
// problem is HBM-bound at ~150MB / 23.3 TB/s, so the f32 matrix pipe's lower
// rate is irrelevant).
//
// Shapes: B=2, S=2048, E=1024, H=16, DH=64.  qkv scratch = 50.3 MB in d_ws.
// ---------------------------------------------------------------------------

typedef __attribute__((ext_vector_type(2))) float v2f;
typedef __attribute__((ext_vector_type(8))) float v8f;

#define BB   2
#define SS   2048
#define EE   1024
#define HH   16
#define DHH  64
#define NE3  3072            // 3*E
#define MM   (BB * SS)       // 4096 rows

// D = A(16x4, f32) * B(4x16, f32) + C(16x16, f32)
__device__ __forceinline__ v8f wmma4(v2f a, v2f b, v8f c) {
    return __builtin_amdgcn_wmma_f32_16x16x4_f32(
        /*neg_a=*/false, a, /*neg_b=*/false, b,
        /*c_mod=*/(short)0, c, /*reuse_a=*/false, /*reuse_b=*/false);
}

// ---------------------------------------------------------------------------
// Kernel 1: qkv[m,f] = sum_e x[m,e] * W[f,e] + bias[f]
//   x   : [M, E]  row-major (B,S flattened)
//   W   : [3E, E] row-major  -> B-matrix element B[k][n] = W[n][k] (contiguous
//                               along k per lane)
// One wave computes a 16x64 output tile (4 accumulators), A reused 4x.
// ---------------------------------------------------------------------------
__global__ __launch_bounds__(256) void qkv_gemm_kernel(
    const float* __restrict__ x, const float* __restrict__ W,
    const float* __restrict__ bias, float* __restrict__ qkv)
{
    const int lane = threadIdx.x & 31;
    const int wave = threadIdx.x >> 5;
    const int l16  = lane & 15;
    const int g    = lane >> 4;          // half-wave group (K pair select)

    const int NT    = NE3 / 64;          // 48 N-tiles
    const int tile  = blockIdx.x * 8 + wave;
    const int mtile = tile / NT;
    const int ntile = tile % NT;
    const int m0    = mtile * 16;
    const int n0    = ntile * 64;

    v8f acc[4] = {{}, {}, {}, {}};

    const float* arow = x + (size_t)(m0 + l16) * EE;

    for (int k = 0; k < EE; k += 4) {
        v2f a;
        a.x = arow[k + 2 * g];
        a.y = arow[k + 2 * g + 1];
#pragma unroll
        for (int t = 0; t < 4; ++t) {
            const float* wr = W + (size_t)(n0 + 16 * t + l16) * EE;
            v2f b;
            b.x = wr[k + 2 * g];
            b.y = wr[k + 2 * g + 1];
            acc[t] = wmma4(a, b, acc[t]);
        }
    }

    // C layout: VGPR r -> row (8*g + r), lane column l16
    const int mrow = m0 + 8 * g;
#pragma unroll
    for (int t = 0; t < 4; ++t) {
        const int col = n0 + 16 * t + l16;
        const float bv = bias[col];
#pragma unroll
        for (int r = 0; r < 8; ++r) {
            qkv[(size_t)(mrow + r) * NE3 + col] = acc[t][r] + bv;
        }
    }
}

// ---------------------------------------------------------------------------
// Kernel 2: flash attention, fp32 WMMA throughout.
// One wave handles one (b, h, 16-query-row tile); 4 waves per block.
//   qkv row s of batch b: base + s*3E;  Q at +0, K at +E, V at +2E, head h at
//   +h*DH within each.
// ---------------------------------------------------------------------------
__global__ __launch_bounds__(128) void attn_kernel(
    const float* __restrict__ qkv, float* __restrict__ out)
{
    __shared__ float pshare[4][16 * 16];   // per-wave P-tile staging (4 KB)

    const int lane  = threadIdx.x & 31;
    const int wave  = threadIdx.x >> 5;
    const int l16   = lane & 15;
    const int g     = lane >> 4;

    const int QT    = SS / 16;             // 128 query tiles
    const int gwave = blockIdx.x * 4 + wave;
    const int qt    = gwave % QT;
    const int bh    = gwave / QT;
    const int h     = bh % HH;
    const int b     = bh / HH;
    const int q0    = qt * 16;

    const float* Qp = qkv + (size_t)b * SS * NE3 + 0 * EE + h * DHH;
    const float* Kp = qkv + (size_t)b * SS * NE3 + 1 * EE + h * DHH;
    const float* Vp = qkv + (size_t)b * SS * NE3 + 2 * EE + h * DHH;

    // Load Q tile into A-fragments: 16 chunks of K=4 over DH=64.
    v2f aq[16];
    {
        const float* qrow = Qp + (size_t)(q0 + l16) * NE3;
#pragma unroll
        for (int kc = 0; kc < 16; ++kc) {
            aq[kc].x = qrow[4 * kc + 2 * g];
            aq[kc].y = qrow[4 * kc + 2 * g + 1];
        }
    }

    v8f o[4] = {{}, {}, {}, {}};           // O tile 16x64
    float mrow[8], lrow[8];
#pragma unroll
    for (int r = 0; r < 8; ++r) { mrow[r] = -3.0e38f; lrow[r] = 0.0f; }

    const float scale = 0.125f;            // 1/sqrt(64)
    float* P = pshare[wave];

    for (int kt = 0; kt < SS / 16; ++kt) {
        const int kbase = kt * 16;

        // ---- scores S = Q * K^T (16x16), B[k][n] = K[key n][dim k] ----
        v8f s = {};
        {
            const float* krow = Kp + (size_t)(kbase + l16) * NE3;
            // prefetch next K/V tiles into cache hierarchy
            if (kt + 1 < SS / 16) {
                __builtin_prefetch(Kp + (size_t)(kbase + 16 + l16) * NE3, 0, 0);
                __builtin_prefetch(Vp + (size_t)(kbase + 16 + l16) * NE3, 0, 0);
            }
#pragma unroll
            for (int kc = 0; kc < 16; ++kc) {
                v2f bk;
                bk.x = krow[4 * kc + 2 * g];
                bk.y = krow[4 * kc + 2 * g + 1];
                s = wmma4(aq[kc], bk, s);
            }
        }

        // ---- online softmax: row stats live across the 16 lanes holding a
        //      row; xor-shuffles with masks 1,2,4,8 stay inside each half ----
        float alpha[8], p[8];
#pragma unroll
        for (int r = 0; r < 8; ++r) {
            float v = s[r] * scale;
            float rm = v;
            rm = fmaxf(rm, __shfl_xor(rm, 1, 32));
            rm = fmaxf(rm, __shfl_xor(rm, 2, 32));
            rm = fmaxf(rm, __shfl_xor(rm, 4, 32));
            rm = fmaxf(rm, __shfl_xor(rm, 8, 32));
            const float mnew = fmaxf(mrow[r], rm);
            const float pe = __expf(v - mnew);
            float rs = pe;
            rs += __shfl_xor(rs, 1, 32);
            rs += __shfl_xor(rs, 2, 32);
            rs += __shfl_xor(rs, 4, 32);
            rs += __shfl_xor(rs, 8, 32);
            alpha[r] = __expf(mrow[r] - mnew);
            lrow[r]  = lrow[r] * alpha[r] + rs;
            mrow[r]  = mnew;
            p[r]     = pe;
        }

        // rescale running O
#pragma unroll
        for (int t = 0; t < 4; ++t)
#pragma unroll
            for (int r = 0; r < 8; ++r)
                o[t][r] *= alpha[r];

        // ---- stage P (C-layout) -> LDS -> reload in A-layout ----
#pragma unroll
        for (int r = 0; r < 8; ++r)
            P[(8 * g + r) * 16 + l16] = p[r];
        __builtin_amdgcn_wave_barrier();
        asm volatile("s_wait_dscnt 0" ::: "memory");

        // ---- O += P * V : A = P chunk (16x4), B = V rows (4x16 per dim tile)
#pragma unroll
        for (int kc = 0; kc < 4; ++kc) {
            v2f ap;
            ap.x = P[l16 * 16 + 4 * kc + 2 * g];
            ap.y = P[l16 * 16 + 4 * kc + 2 * g + 1];
#pragma unroll
            for (int t = 0; t < 4; ++t) {
                const float* vr = Vp + (size_t)(kbase + 4 * kc + 2 * g) * NE3 + 16 * t;
                v2f bv;
                bv.x = vr[l16];          // key 4kc+2g
                bv.y = vr[NE3 + l16];    // key 4kc+2g+1
                o[t] = wmma4(ap, bv, o[t]);
            }
        }
        __builtin_amdgcn_wave_barrier();
        asm volatile("s_wait_dscnt 0" ::: "memory");
    }

    // ---- epilogue: out[b, q0+m, h*DH + d] = O / l ----
    float invl[8];
#pragma unroll
    for (int r = 0; r < 8; ++r) invl[r] = 1.0f / lrow[r];
#pragma unroll
    for (int t = 0; t < 4; ++t) {
#pragma unroll
        for (int r = 0; r < 8; ++r) {
            const int m = 8 * g + r;
            out[((size_t)b * SS + q0 + m) * EE + h * DHH + 16 * t + l16] =
                o[t][r] * invl[r];
        }
    }
}

// ---------------------------------------------------------------------------
extern "C" void kernel_launch(void* const* d_in, const int* in_sizes, int n_in,
                              void* d_out, int out_size, void* d_ws, size_t ws_size,
                              hipStream_t stream) {
    const float* x    = (const float*)d_in[0];   // [B,S,E]
    const float* Wqkv = (const float*)d_in[1];   // [3E,E]
    const float* bqkv = (const float*)d_in[2];   // [3E]
    float* out = (float*)d_out;                  // [B,S,E]
    float* qkv = (float*)d_ws;                   // [B,S,3E] = 50.3 MB scratch

    // 256 M-tiles x 48 N-tiles = 12288 wave-tiles, 8 waves (256 thr) per block
    qkv_gemm_kernel<<<12288 / 8, 256, 0, stream>>>(x, Wqkv, bqkv, qkv);

    // B*H*(S/16) = 4096 waves, 4 waves (128 thr) per block
    attn_kernel<<<4096 / 4, 128, 0, stream>>>(qkv, out);
}